// forward_warp_7335804142212
// MI455X (gfx1250) — compile-verified
//
#include <hip/hip_runtime.h>

// Problem dims from the reference (compile-time constants).
#define BB 8
#define CC 3
#define HH 512
#define WW 1024
#define HW (HH * WW)

// Native (clang ext_vector) float2 so __builtin_nontemporal_load accepts it.
typedef float nfloat2 __attribute__((ext_vector_type(2)));
typedef float nfloat4 __attribute__((ext_vector_type(4)));

// ---------------------------------------------------------------------------
// Kernel 1: zero the accumulation buffer (output is scatter-added into).
// float4 stores for bandwidth; tail handled scalar.
// ---------------------------------------------------------------------------
__global__ __launch_bounds__(256) void zero_f32_kernel(float* __restrict__ out, int n) {
    int n4 = n >> 2;
    int i  = blockIdx.x * blockDim.x + threadIdx.x;
    nfloat4* out4 = reinterpret_cast<nfloat4*>(out);
    if (i < n4) {
        nfloat4 z = {0.f, 0.f, 0.f, 0.f};
        out4[i] = z;
    }
    // tail (n not divisible by 4) — handled by the first few threads
    int tail_start = n4 << 2;
    int t = tail_start + i;
    if (i < (n - tail_start)) {
        out[t] = 0.f;
    }
}

// ---------------------------------------------------------------------------
// Kernel 2: forward warp (bilinear splat) with hardware FP32 global atomics.
// One thread per source pixel (b, y, x); loops over the 3 channels so the
// flow load + weight computation is amortized 3x. Lanes map to consecutive x
// so the flow/im0 loads are fully coalesced. Scatter targets are near-diagonal
// (flow ~ N(0,1)) so the 12 atomics/pixel hit the 192MB L2 with high locality.
// ---------------------------------------------------------------------------
__global__ __launch_bounds__(256) void forward_warp_kernel(
    const float* __restrict__ im0,    // [B, C, H, W]
    const float* __restrict__ flow,   // [B, H, W, 2]
    float*       __restrict__ out)    // [B, C, H, W], pre-zeroed
{
    int tid = blockIdx.x * blockDim.x + threadIdx.x;   // over B*H*W
    if (tid >= BB * HW) return;

    const int b = tid / HW;
    const int p = tid - b * HW;        // y*W + x
    const int y = p >> 10;             // p / WW   (WW == 1024)
    const int x = p & (WW - 1);        // p % WW

    // flow is read exactly once -> non-temporal 8B load
    const nfloat2* flow2 = reinterpret_cast<const nfloat2*>(flow);
    nfloat2 f = __builtin_nontemporal_load(flow2 + tid);

    // reference: flow = clip(flow, -2W, 2W) BEFORE adding the grid
    const float lim = 2.0f * (float)WW;
    float fx = fminf(fmaxf(f.x, -lim), lim);
    float fy = fminf(fmaxf(f.y, -lim), lim);

    float xs = (float)x + fx;
    float ys = (float)y + fy;
    float xf = floorf(xs);
    float yf = floorf(ys);
    int   xi = (int)xf;
    int   yi = (int)yf;

    // reference keeps a tap set only if the full 2x2 neighborhood is in-bounds;
    // otherwise all 4 weights are zeroed -> skipping is exactly equivalent.
    if (xi < 0 || xi > (WW - 2) || yi < 0 || yi > (HH - 2)) return;

    float ax = xs - xf;                // in [0,1)
    float ay = ys - yf;
    float bx = 1.0f - ax;
    float by = 1.0f - ay;
    float w_nw = bx * by;
    float w_ne = ax * by;
    float w_sw = bx * ay;
    float w_se = ax * ay;

    const int base_in  = b * (CC * HW) + p;              // + c*HW per channel
    const int base_out = b * (CC * HW) + yi * WW + xi;   // NW corner

#pragma unroll
    for (int c = 0; c < CC; ++c) {
        // im0 is read exactly once -> non-temporal (keep L2 for the output)
        float v = __builtin_nontemporal_load(&im0[base_in + c * HW]);
        float* o = out + base_out + c * HW;
#if defined(__HIP_DEVICE_COMPILE__) && __has_builtin(__builtin_amdgcn_global_atomic_fadd_f32)
        // hardware global_atomic_add_f32 (no CAS loop)
        unsafeAtomicAdd(o,          v * w_nw);
        unsafeAtomicAdd(o + 1,      v * w_ne);
        unsafeAtomicAdd(o + WW,     v * w_sw);
        unsafeAtomicAdd(o + WW + 1, v * w_se);
#else
        atomicAdd(o,          v * w_nw);
        atomicAdd(o + 1,      v * w_ne);
        atomicAdd(o + WW,     v * w_sw);
        atomicAdd(o + WW + 1, v * w_se);
#endif
    }
}

// ---------------------------------------------------------------------------
// Launch wrapper
// ---------------------------------------------------------------------------
extern "C" void kernel_launch(void* const* d_in, const int* in_sizes, int n_in,
                              void* d_out, int out_size, void* d_ws, size_t ws_size,
                              hipStream_t stream) {
    (void)in_sizes; (void)n_in; (void)d_ws; (void)ws_size;

    const float* im0  = (const float*)d_in[0];   // [B,C,H,W] f32
    const float* flow = (const float*)d_in[1];   // [B,H,W,2] f32
    float*       out  = (float*)     d_out;      // [B,C,H,W] f32

    // 1) zero the accumulator (required every call: harness poisons d_out once
    //    and does not re-poison between graph replays)
    {
        int n4     = out_size >> 2;
        int blocks = (n4 + 255) / 256;
        if (blocks < 1) blocks = 1;
        zero_f32_kernel<<<blocks, 256, 0, stream>>>(out, out_size);
    }

    // 2) scatter splat: one thread per source pixel
    {
        const int npix   = BB * HW;                 // 4,194,304
        const int blocks = (npix + 255) / 256;      // 16,384
        forward_warp_kernel<<<blocks, 256, 0, stream>>>(im0, flow, out);
    }
}